// VectorQuantizerEMA_46514495816277
// MI455X (gfx1250) — compile-verified
//
#include <hip/hip_runtime.h>

typedef __attribute__((ext_vector_type(2))) float v2f;
typedef __attribute__((ext_vector_type(8))) float v8f;
typedef __attribute__((address_space(3))) float lds_f;

#define VQ_D 256
#define VQ_K 4096
#define ROWS_WG 64        // rows (tokens) per workgroup: 4 row-tiles of 16
#define COLS_TILE 32      // codebook columns staged in LDS per buffer
#define EMA_DECAY_F 0.99f
#define ONE_MINUS_F 0.01f
#define LOSS_SCALE_F 10.0f
#define COMMIT_COEF_F 0.25f
#define VQ_EPS_F 1e-5f

#if __has_builtin(__builtin_amdgcn_sched_barrier)
#define SCHED_FENCE() __builtin_amdgcn_sched_barrier(0)
#else
#define SCHED_FENCE() asm volatile("" ::: "memory")
#endif

// Async global -> LDS copy, 16 bytes per lane, tracked by ASYNCcnt.
#define ASYNC_CP_B128(ldsp, gp)                                       \
  asm volatile("global_load_async_to_lds_b128 %0, %1, off" ::"v"(ldsp), \
               "v"(gp)                                                 \
               : "memory")

#if __has_builtin(__builtin_amdgcn_s_wait_asynccnt)
#define ASYNC_WAIT0() __builtin_amdgcn_s_wait_asynccnt(0)
#else
#define ASYNC_WAIT0() asm volatile("s_wait_asynccnt 0x0" ::: "memory")
#endif

// ---------------- zero-init accumulators ----------------
__global__ __launch_bounds__(256) void vq_init_kernel(float* __restrict__ n_acc,
                                                      float* __restrict__ dw_acc,
                                                      float* __restrict__ scal) {
  size_t i = (size_t)blockIdx.x * 256 + threadIdx.x;
  if (i < (size_t)VQ_K * VQ_D) dw_acc[i] = 0.0f;
  if (i < VQ_K) n_acc[i] = 0.0f;
  if (i < 4) scal[i] = 0.0f;
}

// ---------------- per-codebook-row squared norms ----------------
__global__ __launch_bounds__(256) void vq_cnorm_kernel(const float* __restrict__ cb,
                                                       float* __restrict__ cnorm) {
  __shared__ float red[256];
  const int k = blockIdx.x;
  const float v = cb[(size_t)k * VQ_D + threadIdx.x];
  red[threadIdx.x] = v * v;
  __syncthreads();
  for (int s = 128; s > 0; s >>= 1) {
    if (threadIdx.x < s) red[threadIdx.x] += red[threadIdx.x + s];
    __syncthreads();
  }
  if (threadIdx.x == 0) cnorm[k] = red[0];
}

// ---------------- main: async-LDS double-buffered WMMA distance matmul ----------------
__global__ __launch_bounds__(256)
void vq_main_kernel(
    const float* __restrict__ z, const float* __restrict__ mask,
    const float* __restrict__ cb, const float* __restrict__ cnorm,
    float* __restrict__ out, float* __restrict__ idx_out,
    float* __restrict__ n_acc, float* __restrict__ dw_acc,
    float* __restrict__ scal) {
  __shared__ float sB0[COLS_TILE * VQ_D];    // codebook tile buffer 0 (32 KB)
  __shared__ float sB1[COLS_TILE * VQ_D];    // codebook tile buffer 1 (32 KB)

  // Post-loop overlays (both buffers are dead after the final barrier).
  float* sScore = sB0;                         // ROWS_WG*32 floats
  int*   sIdx   = (int*)(sB0 + ROWS_WG * 32);  // ROWS_WG*32 ints
  int*   sBest  = (int*)sB1;                   // ROWS_WG ints
  float* sMask  = sB1 + ROWS_WG;               // ROWS_WG floats
  float* sRed   = sB1 + 2 * ROWS_WG;           // 256 floats

  const int tid   = threadIdx.x;
  const int wave  = tid >> 5;
  const int lane  = tid & 31;
  const int ln    = lane & 15;
  const int half2 = (lane < 16) ? 0 : 2;  // K-offset, ISA 16x4 fp32 A/B layout
  const int rtile   = wave & 3;           // row tile 0..3 (16 rows each)
  const int cstripe = wave >> 2;          // col stripe 0..1 (16 cols each)
  const int rowBase = blockIdx.x * ROWS_WG;

  // B staging assignment: thread -> (col tc, 32-float segment seg); 8 async b128s.
  const int tc  = tid >> 3;
  const int seg = tid & 7;
  const size_t stgOff = (size_t)tc * VQ_D + seg * 32;  // float offset within a tile
  lds_f* l0 = (lds_f*)(sB0 + tc * VQ_D + seg * 32);
  lds_f* l1 = (lds_f*)(sB1 + tc * VQ_D + seg * 32);

  // Async-stage tile 0 into sB0 (no staging registers, ASYNCcnt-tracked).
  {
    const float* g = cb + stgOff;
#pragma unroll
    for (int u = 0; u < 8; ++u) ASYNC_CP_B128(l0 + 4 * u, g + 4 * u);
  }

  // A fragments pinned in registers: lane ln owns row (rtile*16+ln), all 64 K-steps.
  v2f a[64];
  {
    const float* ap = z + (size_t)(rowBase + rtile * 16 + ln) * VQ_D + half2;
#pragma unroll
    for (int t = 0; t < 64; ++t) a[t] = *(const v2f*)(ap + 4 * t);
  }
  ASYNC_WAIT0();
  __syncthreads();

  float best[8];
  int   bidx[8];
#pragma unroll
  for (int r = 0; r < 8; ++r) { best[r] = 3.0e38f; bidx[r] = 0; }

  const float* bp0 = &sB0[(cstripe * 16 + ln) * VQ_D + half2];
  const float* bp1 = &sB1[(cstripe * 16 + ln) * VQ_D + half2];

  const int nTiles = VQ_K / COLS_TILE;  // 128 (even)

  for (int tile = 0; tile < nTiles; tile += 2) {
    // ---------- phase A: compute on sB0 (cols of `tile`), async-stage tile+1 -> sB1
    {
      const float* g = cb + (size_t)(tile + 1) * COLS_TILE * VQ_D + stgOff;
#pragma unroll
      for (int u = 0; u < 8; ++u) ASYNC_CP_B128(l1 + 4 * u, g + 4 * u);
    }
    SCHED_FENCE();  // async copies issue before the WMMA stream
    {
      v8f c = {0.f, 0.f, 0.f, 0.f, 0.f, 0.f, 0.f, 0.f};
#pragma unroll
      for (int t = 0; t < 64; ++t) {
        v2f b = *(const v2f*)(bp0 + 4 * t);
        c = __builtin_amdgcn_wmma_f32_16x16x4_f32(false, a[t], false, b,
                                                  (short)0, c, false, false);
      }
      const int col = tile * COLS_TILE + cstripe * 16 + ln;
      const float cn = cnorm[col];
#pragma unroll
      for (int rr = 0; rr < 8; ++rr) {
        const float s = cn - 2.0f * c[rr];
        if (s < best[rr]) { best[rr] = s; bidx[rr] = col; }
      }
    }
    ASYNC_WAIT0();   // tile+1 fully landed in sB1
    __syncthreads();

    // ---------- phase B: compute on sB1 (cols of tile+1), async-stage tile+2 -> sB0
    if (tile + 2 < nTiles) {
      const float* g = cb + (size_t)(tile + 2) * COLS_TILE * VQ_D + stgOff;
#pragma unroll
      for (int u = 0; u < 8; ++u) ASYNC_CP_B128(l0 + 4 * u, g + 4 * u);
    }
    SCHED_FENCE();
    {
      v8f c = {0.f, 0.f, 0.f, 0.f, 0.f, 0.f, 0.f, 0.f};
#pragma unroll
      for (int t = 0; t < 64; ++t) {
        v2f b = *(const v2f*)(bp1 + 4 * t);
        c = __builtin_amdgcn_wmma_f32_16x16x4_f32(false, a[t], false, b,
                                                  (short)0, c, false, false);
      }
      const int col = (tile + 1) * COLS_TILE + cstripe * 16 + ln;
      const float cn = cnorm[col];
#pragma unroll
      for (int rr = 0; rr < 8; ++rr) {
        const float s = cn - 2.0f * c[rr];
        if (s < best[rr]) { best[rr] = s; bidx[rr] = col; }
      }
    }
    ASYNC_WAIT0();
    __syncthreads();
  }

  // Dump per-lane running mins into the overlay (buffers dead now).
  // C layout: reg r -> row r (lanes 0-15) / r+8 (lanes 16-31)
  {
    const int rowOff = (lane < 16) ? 0 : 8;
    const int slot = cstripe * 16 + ln;
#pragma unroll
    for (int r = 0; r < 8; ++r) {
      const int lrow = rtile * 16 + r + rowOff;
      sScore[lrow * 32 + slot] = best[r];
      sIdx[lrow * 32 + slot]   = bidx[r];
    }
  }
  if (tid < ROWS_WG) sMask[tid] = mask[rowBase + tid];
  __syncthreads();

  // Final argmin per row (tie -> lowest index, matching jnp.argmin)
  if (tid < ROWS_WG) {
    float bs = 3.0e38f;
    int bi = VQ_K;
    for (int s = 0; s < 32; ++s) {
      const float v = sScore[tid * 32 + s];
      const int  ii = sIdx[tid * 32 + s];
      if (v < bs || (v == bs && ii < bi)) { bs = v; bi = ii; }
    }
    sBest[tid] = bi;
    idx_out[rowBase + tid] = (float)bi;
  }
  __syncthreads();

  // Fused epilogue: quantized output, loss partial, EMA scatter (dw, n)
  float lpart = 0.0f;
  for (int m = 0; m < ROWS_WG; ++m) {
    const int k = sBest[m];
    const float q  = cb[(size_t)k * VQ_D + tid];
    const float zz = z[(size_t)(rowBase + m) * VQ_D + tid];
    const float mk = sMask[m];
    out[(size_t)(rowBase + m) * VQ_D + tid] = q;  // z + sg(q - z) == q numerically
    const float d = zz - q;
    lpart += d * d * mk;
    atomicAdd(&dw_acc[(size_t)k * VQ_D + tid], zz * mk);
  }
  if (tid < ROWS_WG) atomicAdd(&n_acc[sBest[tid]], sMask[tid]);

  sRed[tid] = lpart;
  __syncthreads();
  for (int s = 128; s > 0; s >>= 1) {
    if (tid < s) sRed[tid] += sRed[tid + s];
    __syncthreads();
  }
  if (tid == 0) {
    atomicAdd(&scal[0], sRed[0]);  // sum (z-q)^2 * mask
    float ms = 0.0f;
    for (int m = 0; m < ROWS_WG; ++m) ms += sMask[m];
    atomicAdd(&scal[1], ms);       // sum mask
  }
}

// ---------------- count EMA + global total ----------------
__global__ __launch_bounds__(256) void vq_count_kernel(const float* __restrict__ ema_count,
                                                       const float* __restrict__ n_acc,
                                                       float* __restrict__ nc,
                                                       float* __restrict__ scal) {
  __shared__ float red[256];
  const int k = blockIdx.x * 256 + threadIdx.x;
  const float v = ema_count[k] * EMA_DECAY_F + ONE_MINUS_F * n_acc[k];
  nc[k] = v;
  red[threadIdx.x] = v;
  __syncthreads();
  for (int s = 128; s > 0; s >>= 1) {
    if (threadIdx.x < s) red[threadIdx.x] += red[threadIdx.x + s];
    __syncthreads();
  }
  if (threadIdx.x == 0) atomicAdd(&scal[2], red[0]);
}

// ---------------- finalize: new_count (Laplace), new_weight, loss ----------------
__global__ __launch_bounds__(256) void vq_final_kernel(
    const float* __restrict__ nc, const float* __restrict__ ema_weight,
    const float* __restrict__ dw_acc, const float* __restrict__ scal,
    float* __restrict__ loss_out, float* __restrict__ count_out,
    float* __restrict__ weight_out) {
  const size_t i = (size_t)blockIdx.x * 256 + threadIdx.x;
  if (i < (size_t)VQ_K * VQ_D)
    weight_out[i] = ema_weight[i] * EMA_DECAY_F + ONE_MINUS_F * dw_acc[i];
  if (i < VQ_K) {
    const float total = scal[2];
    count_out[i] = (nc[i] + VQ_EPS_F) / (total + (float)VQ_K * VQ_EPS_F) * total;
  }
  if (i == 0)
    loss_out[0] = LOSS_SCALE_F * COMMIT_COEF_F * (scal[0] / (float)VQ_D) / scal[1];
}

// ---------------- launcher ----------------
extern "C" void kernel_launch(void* const* d_in, const int* in_sizes, int n_in,
                              void* d_out, int out_size, void* d_ws, size_t ws_size,
                              hipStream_t stream) {
  const float* z          = (const float*)d_in[0];  // [B,S,D]
  const float* mask       = (const float*)d_in[1];  // [B,S]
  const float* cb         = (const float*)d_in[2];  // [K,D]
  const float* ema_count  = (const float*)d_in[3];  // [K]
  const float* ema_weight = (const float*)d_in[4];  // [K,D]

  const int N = in_sizes[1];  // B*S tokens (16384); D=256, K=4096 fixed

  float* outp       = (float*)d_out;                  // [N,D]
  float* idx_out    = outp + (size_t)N * VQ_D;        // [N]
  float* loss_out   = idx_out + N;                    // [1]
  float* count_out  = loss_out + 1;                   // [K]
  float* weight_out = count_out + VQ_K;               // [K,D]

  float* ws     = (float*)d_ws;
  float* cnorm  = ws;                 // K
  float* n_acc  = cnorm + VQ_K;       // K
  float* nc     = n_acc + VQ_K;       // K
  float* scal   = nc + VQ_K;          // 4 scalars: loss_sum, mask_sum, total
  float* dw_acc = scal + 4;           // K*D

  const int kd_blocks = (VQ_K * VQ_D + 255) / 256;

  vq_init_kernel<<<kd_blocks, 256, 0, stream>>>(n_acc, dw_acc, scal);
  vq_cnorm_kernel<<<VQ_K, 256, 0, stream>>>(cb, cnorm);
  vq_main_kernel<<<N / ROWS_WG, 256, 0, stream>>>(z, mask, cb, cnorm, outp,
                                                  idx_out, n_acc, dw_acc, scal);
  vq_count_kernel<<<VQ_K / 256, 256, 0, stream>>>(ema_count, n_acc, nc, scal);
  vq_final_kernel<<<kd_blocks, 256, 0, stream>>>(nc, ema_weight, dw_acc, scal,
                                                 loss_out, count_out, weight_out);
}